// YoloEval_62130996904475
// MI455X (gfx1250) — compile-verified
//
#include <hip/hip_runtime.h>
#include <hip/hip_bf16.h>
#include <stdint.h>

// ---------------------------------------------------------------------------
// YOLO eval post-processing for MI455X (gfx1250).
// K1 decode (TDM-staged, elementwise sigmoid/exp) ->
// K2 per-class wave32-ballot compaction (TDM double-buffered score stream) ->
// K3 per-class greedy NMS in LDS (TDM-loaded candidate blob).
// ---------------------------------------------------------------------------

#define NUM_CLASSES 80
#define MAX_BOXES   20
#define SCORE_THR   0.6f
#define IOU_THR     0.5f
#define IN_DIM      1664.0f           // 52*32 network input (square)

#define N0 8112                        // 52*52*3
#define N1 32448                       // 104*104*3
#define N2 129792                      // 208*208*3
#define NTOT (N0 + N1 + N2)            // 170352 boxes

#define KCAND 2048                     // per-class candidate cap (LDS-resident)
#define CHUNK 2048                     // score-stream chunk (floats) per TDM load
#define TILE_BOXES 128                 // boxes per decode block (128*85*4B = 42.5KB LDS)

#define SLOTS (NUM_CLASSES * MAX_BOXES)   // 1600
#define OUT_SCORES  (SLOTS * 4)           // 6400
#define OUT_CLASSES (OUT_SCORES + SLOTS)  // 8000
#define OUT_VALID   (OUT_CLASSES + SLOTS) // 9600  (total 11200 floats)

// ---------------- CDNA5 Tensor Data Mover (TDM) ----------------------------
#if defined(__has_builtin)
#if __has_builtin(__builtin_amdgcn_tensor_load_to_lds)
#define HAVE_TDM 1
#endif
#endif

#if defined(__has_include)
#if __has_include(<hip/amd_detail/amd_gfx1250_TDM.h>)
#define TDM_SIX_ARGS 1   // therock-10.0 headers => 6-arg builtin
#endif
#endif

typedef unsigned int u32x4 __attribute__((ext_vector_type(4)));
typedef int          i32x8 __attribute__((ext_vector_type(8)));
typedef int          i32x4 __attribute__((ext_vector_type(4)));

#if defined(HAVE_TDM)
// 1-D contiguous fp32 DMA: global -> LDS.  tensor_elems bounds the read so the
// TDM zero-fills past the end of the tensor (free tail handling).
// D# layout per CDNA5 ISA ch.8: group0 = {count|flags, lds_addr, gaddr_lo,
// gaddr_hi|type=2}; group1 word0 data_size=4B, tensor_dim0 @bits[79:48],
// tensor_dim1=1 @[111:80], tile_dim0 @[127:112], tile_dim1/2 = 0 (1-D).
__device__ __forceinline__ void tdm_load_f32_1d(void* lds_dst, const void* gsrc,
                                                unsigned tile_elems,
                                                unsigned long long tensor_elems) {
  unsigned long long ga = (unsigned long long)(uintptr_t)gsrc;
  u32x4 g0;
  g0[0] = 1u;                                                   // count=1, user mode
  g0[1] = (unsigned)(uintptr_t)lds_dst;                         // LDS byte offset (low 32b of generic LDS ptr)
  g0[2] = (unsigned)ga;                                         // global_addr[31:0]
  g0[3] = (unsigned)((ga >> 32) & 0x1FFFFFFull) | (2u << 30);   // global_addr[56:32] | type=2
  unsigned td0 = (tensor_elems > 0xFFFFFFFFull) ? 0xFFFFFFFFu : (unsigned)tensor_elems;
  i32x8 g1;
  g1[0] = (int)(2u << 16);                     // data_size = 2 -> 4 bytes; no pad/iter/multicast
  g1[1] = (int)((td0 & 0xFFFFu) << 16);        // tensor_dim0[15:0]
  g1[2] = (int)((td0 >> 16) | (1u << 16));     // tensor_dim0[31:16] | tensor_dim1 = 1
  g1[3] = (int)((tile_elems & 0xFFFFu) << 16); // tile_dim0
  g1[4] = 0;                                   // tile_dim1 = 0 (unused), tile_dim2 = 0
  g1[5] = (int)tile_elems;                     // tensor_dim0_stride (don't-care for 1-D)
  g1[6] = 0;
  g1[7] = 0;
  i32x4 z4 = {0, 0, 0, 0};
#if defined(TDM_SIX_ARGS)
  i32x8 z8 = {0, 0, 0, 0, 0, 0, 0, 0};
  __builtin_amdgcn_tensor_load_to_lds(g0, g1, z4, z4, z8, 0);
#else
  __builtin_amdgcn_tensor_load_to_lds(g0, g1, z4, z4, 0);
#endif
}
#endif // HAVE_TDM

__device__ __forceinline__ float sigmoidf_(float x) { return 1.0f / (1.0f + __expf(-x)); }

// ---------------- K1: decode one scale -------------------------------------
__global__ __launch_bounds__(TILE_BOXES)
void decode_kernel(const float* __restrict__ feat,
                   const float* __restrict__ image_shape,
                   const float* __restrict__ anchors,
                   float* __restrict__ d_boxes,     // [NTOT][4]  (y1,x1,y2,x2)
                   float* __restrict__ d_scoresT,   // [80][NTOT]
                   int S, int nscale, int scale_base, int anchor_row0) {
  __shared__ float tile[TILE_BOXES * 85];
  const int blockStart = blockIdx.x * TILE_BOXES;

#if defined(HAVE_TDM)
  if ((threadIdx.x >> 5) == 0) {   // wave 0 issues the DMA; EXEC ignored by TDM
    tdm_load_f32_1d(tile, feat + (size_t)blockStart * 85, TILE_BOXES * 85,
                    (unsigned long long)(nscale - blockStart) * 85ull);
    __builtin_amdgcn_s_wait_tensorcnt(0);
  }
#else
  for (int i = threadIdx.x; i < TILE_BOXES * 85; i += blockDim.x) {
    long g = (long)blockStart * 85 + i;
    tile[i] = (g < (long)nscale * 85) ? feat[g] : 0.0f;
  }
#endif
  __syncthreads();

  const int t = threadIdx.x;
  const int n = blockStart + t;
  if (n >= nscale) return;

  const float* f = &tile[t * 85];
  const int a    = n % 3;
  const int cell = n / 3;
  const int gx = cell % S;
  const int gy = cell / S;

  // yolo_correct_boxes affine (general form, reads image_shape from input)
  const float ih = image_shape[0], iw = image_shape[1];
  const float mx = fmaxf(ih, iw);
  const float boxedh = IN_DIM * (ih / mx), boxedw = IN_DIM * (iw / mx);
  const float offh = (IN_DIM - boxedh) * 0.5f, offw = (IN_DIM - boxedw) * 0.5f;
  const float sch = ih / boxedh, scw = iw / boxedw;

  const float aw = anchors[2 * (anchor_row0 + a) + 0];
  const float ah = anchors[2 * (anchor_row0 + a) + 1];

  const float invS = 1.0f / (float)S;
  const float bx = (sigmoidf_(f[0]) + (float)gx) * invS;
  const float by = (sigmoidf_(f[1]) + (float)gy) * invS;
  const float bw = __expf(f[2]) * aw * (1.0f / IN_DIM);
  const float bh = __expf(f[3]) * ah * (1.0f / IN_DIM);
  const float conf = sigmoidf_(f[4]);

  const float ypix = (by * IN_DIM - offh) * sch;
  const float xpix = (bx * IN_DIM - offw) * scw;
  const float hpix = bh * IN_DIM * sch;
  const float wpix = bw * IN_DIM * scw;

  const float y1 = fminf(fmaxf(ypix - 0.5f * hpix, 0.0f), ih);
  const float x1 = fminf(fmaxf(xpix - 0.5f * wpix, 0.0f), iw);
  const float y2 = fminf(fmaxf(ypix + 0.5f * hpix, 0.0f), ih);
  const float x2 = fminf(fmaxf(xpix + 0.5f * wpix, 0.0f), iw);

  const int gn = scale_base + n;
  reinterpret_cast<float4*>(d_boxes)[gn] = make_float4(y1, x1, y2, x2);

#pragma unroll 8
  for (int c = 0; c < NUM_CLASSES; ++c)
    d_scoresT[(size_t)c * NTOT + gn] = conf * sigmoidf_(f[5 + c]);
}

// ---------------- K2: per-class ordered compaction -------------------------
// One block per class. TDM double-buffers 8KB score chunks (L2-resident after
// K1); wave32 ballot + popc gives a deterministic, order-preserving compaction
// of candidates >= SCORE_THR into a per-class blob [score K][idx K][box 4K].
__global__ __launch_bounds__(256)
void compact_kernel(const float* __restrict__ d_scoresT,
                    const float* __restrict__ d_boxes,
                    float* __restrict__ d_cand,
                    int* __restrict__ d_counts) {
  __shared__ float buf[2][CHUNK];
  __shared__ unsigned wcnt[8];

  const int c = blockIdx.x;
  const int tid = threadIdx.x, lane = tid & 31, wid = tid >> 5;
  const float* row = d_scoresT + (size_t)c * NTOT;
  float* candS = d_cand + (size_t)c * (6 * KCAND);
  int*   candI = (int*)(candS + KCAND);
  float* candB = candS + 2 * KCAND;

  const int nchunks = (NTOT + CHUNK - 1) / CHUNK;
  unsigned running = 0;

#if defined(HAVE_TDM)
  if (wid == 0)
    tdm_load_f32_1d(buf[0], row, CHUNK, (unsigned long long)NTOT);
#endif

  for (int ch = 0; ch < nchunks; ++ch) {
    const int start = ch * CHUNK;
#if defined(HAVE_TDM)
    if (wid == 0) {
      if (ch + 1 < nchunks) {
        tdm_load_f32_1d(buf[(ch + 1) & 1], row + (size_t)(ch + 1) * CHUNK, CHUNK,
                        (unsigned long long)(NTOT - (ch + 1) * CHUNK));
        __builtin_amdgcn_s_wait_tensorcnt(1);   // chunk ch landed
      } else {
        __builtin_amdgcn_s_wait_tensorcnt(0);
      }
    }
#else
    for (int i = tid; i < CHUNK; i += 256) {
      int e = start + i;
      buf[ch & 1][i] = (e < NTOT) ? row[e] : -1.0f;
    }
#endif
    __syncthreads();
    const float* b = buf[ch & 1];

    for (int j = 0; j < CHUNK / 256; ++j) {
      const int e = start + j * 256 + tid;
      const float s = b[j * 256 + tid];
      const bool pred = (e < NTOT) && (s >= SCORE_THR);
      const unsigned m = (unsigned)__ballot(pred);          // wave32 mask
      const unsigned prefix = __popc(m & ((1u << lane) - 1u));
      if (lane == 0) wcnt[wid] = (unsigned)__popc(m);
      __syncthreads();
      unsigned waveoff = 0, tot = 0;
      for (int w = 0; w < 8; ++w) {
        unsigned v = wcnt[w];
        if (w < wid) waveoff += v;
        tot += v;
      }
      const unsigned off = running + waveoff + prefix;
      if (pred && off < KCAND) {
        candS[off] = s;
        candI[off] = e;
        float4 bb = reinterpret_cast<const float4*>(d_boxes)[e];
        candB[4 * off + 0] = bb.x; candB[4 * off + 1] = bb.y;
        candB[4 * off + 2] = bb.z; candB[4 * off + 3] = bb.w;
      }
      running += tot;
      __syncthreads();
    }
  }
  if (tid == 0) d_counts[c] = (int)(running < KCAND ? running : KCAND);
}

// ---------------- K3: per-class greedy NMS in LDS --------------------------
__global__ __launch_bounds__(256)
void nms_kernel(const float* __restrict__ d_cand,
                const int* __restrict__ d_counts,
                const float* __restrict__ d_boxes,
                float* __restrict__ out) {
  __shared__ float blob[6 * KCAND];   // [score K][idx K][box 4K] = 48KB
  __shared__ float rv[256];
  __shared__ int   ri[256];
  __shared__ float sel[4];
  __shared__ int   svalid;

  const int c = blockIdx.x, tid = threadIdx.x, wid = tid >> 5;
  const float* src = d_cand + (size_t)c * (6 * KCAND);

#if defined(HAVE_TDM)
  if (wid == 0) {
    tdm_load_f32_1d(blob, src, 6 * KCAND, (unsigned long long)(6 * KCAND));
    __builtin_amdgcn_s_wait_tensorcnt(0);
  }
#else
  for (int i = tid; i < 6 * KCAND; i += 256) blob[i] = src[i];
#endif
  __syncthreads();

  const int cnt = d_counts[c];
  float* sc  = blob;
  float* bxs = blob + 2 * KCAND;
  const float NEG = -__builtin_inff();

  for (int k = tid; k < KCAND; k += 256)
    if (k >= cnt) sc[k] = NEG;
  __syncthreads();

  for (int it = 0; it < MAX_BOXES; ++it) {
    // block argmax, tie-break lowest index (== jnp.argmax first-max rule,
    // candidate order preserves original box order from K2)
    float v = NEG; int bi = 0x7fffffff;
    for (int k = tid; k < KCAND; k += 256) {
      float s = sc[k];
      if (s > v || (s == v && k < bi)) { v = s; bi = k; }
    }
    rv[tid] = v; ri[tid] = bi;
    __syncthreads();
    for (int st = 128; st > 0; st >>= 1) {
      if (tid < st) {
        float ov = rv[tid + st]; int oi = ri[tid + st];
        if (ov > rv[tid] || (ov == rv[tid] && oi < ri[tid])) { rv[tid] = ov; ri[tid] = oi; }
      }
      __syncthreads();
    }

    if (tid == 0) {
      const float bv = rv[0];
      const int   bk = ri[0];
      const int valid = (bv > NEG) ? 1 : 0;
      const int slot = c * MAX_BOXES + it;
      float b0, b1, b2, b3;
      if (valid) {
        b0 = bxs[4 * bk + 0]; b1 = bxs[4 * bk + 1];
        b2 = bxs[4 * bk + 2]; b3 = bxs[4 * bk + 3];
        sc[bk] = NEG;                       // .at[i].set(-inf)
      } else {
        // reference: argmax over all -inf -> index 0 -> global boxes[0]
        b0 = d_boxes[0]; b1 = d_boxes[1]; b2 = d_boxes[2]; b3 = d_boxes[3];
      }
      out[slot * 4 + 0] = truncf(b0);       // astype(int32)
      out[slot * 4 + 1] = truncf(b1);
      out[slot * 4 + 2] = truncf(b2);
      out[slot * 4 + 3] = truncf(b3);
      out[OUT_SCORES + slot]  = valid ? bv : 0.0f;
      out[OUT_CLASSES + slot] = (float)c;
      out[OUT_VALID + slot]   = (float)valid;
      sel[0] = b0; sel[1] = b1; sel[2] = b2; sel[3] = b3;
      svalid = valid;
    }
    __syncthreads();

    if (svalid) {                            // reference skips suppression when invalid
      const float sy1 = sel[0], sx1 = sel[1], sy2 = sel[2], sx2 = sel[3];
      const float a1 = fmaxf(sy2 - sy1, 0.0f) * fmaxf(sx2 - sx1, 0.0f);
      for (int k = tid; k < KCAND; k += 256) {
        if (sc[k] == NEG) continue;
        const float y1 = bxs[4 * k + 0], x1 = bxs[4 * k + 1];
        const float y2 = bxs[4 * k + 2], x2 = bxs[4 * k + 3];
        const float ihh = fminf(sy2, y2) - fmaxf(sy1, y1);
        const float iww = fminf(sx2, x2) - fmaxf(sx1, x1);
        const float inter = fmaxf(ihh, 0.0f) * fmaxf(iww, 0.0f);
        const float a2 = fmaxf(y2 - y1, 0.0f) * fmaxf(x2 - x1, 0.0f);
        const float iou = inter / fmaxf(a1 + a2 - inter, 1e-9f);
        if (iou > IOU_THR) sc[k] = NEG;
      }
    }
    __syncthreads();
  }
}

// ---------------- launch ----------------------------------------------------
extern "C" void kernel_launch(void* const* d_in, const int* in_sizes, int n_in,
                              void* d_out, int out_size, void* d_ws, size_t ws_size,
                              hipStream_t stream) {
  const float* feat0 = (const float*)d_in[0];
  const float* feat1 = (const float*)d_in[1];
  const float* feat2 = (const float*)d_in[2];
  const float* img   = (const float*)d_in[3];
  const float* anch  = (const float*)d_in[4];
  float* out = (float*)d_out;

  // workspace layout (floats): boxes | scoresT | counts(pad 128 ints) | cand
  float* ws        = (float*)d_ws;
  float* d_boxes   = ws;
  float* d_scoresT = ws + (size_t)NTOT * 4;
  int*   d_counts  = (int*)(d_scoresT + (size_t)NUM_CLASSES * NTOT);
  float* d_cand    = (float*)(d_counts + 128);

  decode_kernel<<<(N0 + TILE_BOXES - 1) / TILE_BOXES, TILE_BOXES, 0, stream>>>(
      feat0, img, anch, d_boxes, d_scoresT, 52, N0, 0, 6);
  decode_kernel<<<(N1 + TILE_BOXES - 1) / TILE_BOXES, TILE_BOXES, 0, stream>>>(
      feat1, img, anch, d_boxes, d_scoresT, 104, N1, N0, 3);
  decode_kernel<<<(N2 + TILE_BOXES - 1) / TILE_BOXES, TILE_BOXES, 0, stream>>>(
      feat2, img, anch, d_boxes, d_scoresT, 208, N2, N0 + N1, 0);

  compact_kernel<<<NUM_CLASSES, 256, 0, stream>>>(d_scoresT, d_boxes, d_cand, d_counts);
  nms_kernel<<<NUM_CLASSES, 256, 0, stream>>>(d_cand, d_counts, d_boxes, out);
}